// TFMultinomialCloglogEstimator_59828894433694
// MI455X (gfx1250) — compile-verified
//
#include <hip/hip_runtime.h>

// CDNA5 / gfx1250: wave32, WMMA 16x16x4 F32 path for the X@beta matvec.
typedef __attribute__((ext_vector_type(2))) float v2f;
typedef __attribute__((ext_vector_type(8))) float v8f;

#define NFEAT 32
#define NCLASS 8
#define PEPS 1e-9f

// ---------------------------------------------------------------------------
// Kernel 1: sort the (K-1)=7 intercepts into d_ws (ascending), one thread.
// Re-run every launch (d_ws is poisoned by the harness).
// ---------------------------------------------------------------------------
__global__ void sort_intercepts_kernel(const float* __restrict__ intercepts,
                                       float* __restrict__ sorted, int k) {
  if (threadIdx.x == 0 && blockIdx.x == 0) {
    float v[16];
    for (int i = 0; i < k; ++i) v[i] = intercepts[i];
    for (int i = 1; i < k; ++i) {           // insertion sort, k == 7
      float key = v[i];
      int j = i - 1;
      while (j >= 0 && v[j] > key) { v[j + 1] = v[j]; --j; }
      v[j + 1] = key;
    }
    for (int i = 0; i < k; ++i) sorted[i] = v[i];
  }
}

// ---------------------------------------------------------------------------
// Main kernel: one 16-row tile per wave.
//   eta(16 rows) via 8x accumulated V_WMMA_F32_16X16X4_F32:
//     A = 16x32 X tile (lane<16: K=4k..4k+1 of row lane; lane>=16: K=4k+2..4k+3)
//     B = beta broadcast over all 16 N columns -> D[m][n] = eta_m
//   C/D layout: lanes 0..15 hold eta_0..7 in c[0..7], lanes 16..31 hold eta_8..15.
//   Lane l writes out tile elements [4l, 4l+4): row = l>>1, classes (l&1)*4..+3.
// ---------------------------------------------------------------------------
__global__ __launch_bounds__(256) void cloglog_wmma_kernel(
    const float* __restrict__ X, const float* __restrict__ beta,
    const float* __restrict__ th /* sorted, 7 */, float* __restrict__ out,
    int ntiles) {
  const int lane = threadIdx.x & 31;
  const bool hi = lane >= 16;               // upper half-wave: K offset +2
  const int waves_per_block = blockDim.x >> 5;
  const int wave = blockIdx.x * waves_per_block + (threadIdx.x >> 5);
  const int wave_stride = gridDim.x * waves_per_block;

  // Per-lane B operand: beta replicated across N. Chunk k covers K=4k..4k+3.
  // B layout (4x16 f32, mirrors A): VGPR0 = {K=4k | K=4k+2}, VGPR1 = {K=4k+1 | K=4k+3}.
  v2f bvec[8];
#pragma unroll
  for (int k = 0; k < 8; ++k) {
    const int off = k * 4 + (hi ? 2 : 0);
    bvec[k].x = beta[off];
    bvec[k].y = beta[off + 1];
  }

  // Sorted thresholds: uniform -> scalar registers.
  float ths[NCLASS - 1];
#pragma unroll
  for (int j = 0; j < NCLASS - 1; ++j) ths[j] = th[j];

  const int myrow = lane & 15;              // A-matrix M for this lane
  const int koff = hi ? 2 : 0;

  for (int tile = wave; tile < ntiles; tile += wave_stride) {   // uniform cond
    const float* xrow = X + (size_t)(tile * 16 + myrow) * NFEAT + koff;

    v8f c = {};
#pragma unroll
    for (int k = 0; k < 8; ++k) {
      v2f a = *(const v2f*)(xrow + k * 4);  // 8B-aligned float2
      // (neg_a, A, neg_b, B, c_mod, C, reuse_a, reuse_b)
      c = __builtin_amdgcn_wmma_f32_16x16x4_f32(
          false, a, false, bvec[k], (short)0, c, false, false);
    }

    // Extract eta for this lane's output row: r = (lane>>1)&7, select tree.
    const int r = (lane >> 1) & 7;
    float e0 = (r & 1) ? c[1] : c[0];
    float e1 = (r & 1) ? c[3] : c[2];
    float e2 = (r & 1) ? c[5] : c[4];
    float e3 = (r & 1) ? c[7] : c[6];
    float f0 = (r & 2) ? e1 : e0;
    float f1 = (r & 2) ? e3 : e2;
    const float eta = (r & 4) ? f1 : f0;

    // cloglog cumulatives for all 7 thresholds (fully unrolled, v_exp_f32).
    float cum[NCLASS - 1];
#pragma unroll
    for (int j = 0; j < NCLASS - 1; ++j) {
      float lp = fminf(fmaxf(ths[j] - eta, -30.0f), 30.0f);
      cum[j] = 1.0f - __expf(-__expf(lp));
    }

    // Lane covers classes cls0..cls0+3 with cls0 = (lane&1)*4.
    // cumulative[-1] = 0, cumulative[7] = 1.
    const bool odd = (lane & 1) != 0;
    float s0 = odd ? cum[3] : 0.0f;
    float s1 = odd ? cum[4] : cum[0];
    float s2 = odd ? cum[5] : cum[1];
    float s3 = odd ? cum[6] : cum[2];
    float s4 = odd ? 1.0f   : cum[3];

    float4 p;
    p.x = fminf(fmaxf(s1 - s0, PEPS), 1.0f - PEPS);
    p.y = fminf(fmaxf(s2 - s1, PEPS), 1.0f - PEPS);
    p.z = fminf(fmaxf(s3 - s2, PEPS), 1.0f - PEPS);
    p.w = fminf(fmaxf(s4 - s3, PEPS), 1.0f - PEPS);

    // Tile output = 128 consecutive floats; lane l stores [4l,4l+4) -> 512B/wave.
    *(float4*)(out + (size_t)tile * (16 * NCLASS) + lane * 4) = p;
  }
}

// ---------------------------------------------------------------------------
// Scalar tail for rows not covered by a full 16-row tile (N % 16).
// ---------------------------------------------------------------------------
__global__ void cloglog_tail_kernel(const float* __restrict__ X,
                                    const float* __restrict__ beta,
                                    const float* __restrict__ th,
                                    float* __restrict__ out, int row0, int nrows) {
  int r = row0 + blockIdx.x * blockDim.x + threadIdx.x;
  if (r >= nrows) return;
  float eta = 0.0f;
#pragma unroll
  for (int j = 0; j < NFEAT; ++j) eta += X[(size_t)r * NFEAT + j] * beta[j];
  float prev = 0.0f;
#pragma unroll
  for (int cidx = 0; cidx < NCLASS; ++cidx) {
    float cur;
    if (cidx < NCLASS - 1) {
      float lp = fminf(fmaxf(th[cidx] - eta, -30.0f), 30.0f);
      cur = 1.0f - __expf(-__expf(lp));
    } else {
      cur = 1.0f;
    }
    out[(size_t)r * NCLASS + cidx] = fminf(fmaxf(cur - prev, PEPS), 1.0f - PEPS);
    prev = cur;
  }
}

extern "C" void kernel_launch(void* const* d_in, const int* in_sizes, int n_in,
                              void* d_out, int out_size, void* d_ws, size_t ws_size,
                              hipStream_t stream) {
  const float* X          = (const float*)d_in[0];
  const float* intercepts = (const float*)d_in[1];
  const float* beta       = (const float*)d_in[2];
  float* out = (float*)d_out;

  const int kints = in_sizes[1];            // 7
  const int N = in_sizes[0] / NFEAT;        // 2,000,000 rows
  float* sorted = (float*)d_ws;

  sort_intercepts_kernel<<<1, 32, 0, stream>>>(intercepts, sorted, kints);

  const int ntiles = N / 16;
  if (ntiles > 0) {
    const int waves_per_block = 8;          // 256 threads = 8 wave32
    int blocks = (ntiles + waves_per_block - 1) / waves_per_block;
    cloglog_wmma_kernel<<<blocks, 256, 0, stream>>>(X, beta, sorted, out, ntiles);
  }
  const int rem = N % 16;
  if (rem > 0) {
    cloglog_tail_kernel<<<1, 64, 0, stream>>>(X, beta, sorted, out, N - rem, N);
  }
}